// GroundTruthSDE_38439957299704
// MI455X (gfx1250) — compile-verified
//
#include <hip/hip_runtime.h>

typedef __attribute__((ext_vector_type(2))) float v2f;
typedef __attribute__((ext_vector_type(8))) float v8f;

#define N_PART   32768
#define DIM      32
#define KMIX     16
#define NSTEPS   100
#define EPS_F    1.0f
#define DT_F     0.01f
#define NOISE_SC 0.1f           // sqrt(EPS)*sqrt(dt)
#define PAD      33             // LDS row stride (floats) -> conflict-free
#define PPB      128            // particles per block (8 waves * 16)
#define TWO_PI_F 6.28318530717958647692f

// 32-bit avalanche hash (lowbias32) -- cheap deterministic counter RNG
__device__ __forceinline__ unsigned int lowbias32(unsigned int x) {
    x ^= x >> 16; x *= 0x21f0aaadu;
    x ^= x >> 15; x *= 0x735a2d97u;
    x ^= x >> 15;
    return x;
}

__global__ __launch_bounds__(256, 2)
void sde_traj_kernel(const float* __restrict__ x0,
                     const float* __restrict__ probs,
                     const float* __restrict__ mus,
                     const float* __restrict__ sigmas,
                     float* __restrict__ traj)
{
    __shared__ float sInvVar[KMIX * PAD];   // 1/var[k][d]
    __shared__ float sMuInv [KMIX * PAD];   // mu/var[k][d]
    __shared__ float sTerm  [KMIX * PAD];   // mu^2/var + log(var) per (k,d)
    __shared__ float sBias  [KMIX];         // log_w - 0.5*sum_d term (2pi const dropped)
    __shared__ float sLogW  [KMIX];
    __shared__ float sX     [PPB * PAD];    // current particle state

    const int tid    = threadIdx.x;
    const int wave   = tid >> 5;
    const int lane   = tid & 31;
    const int half   = lane >> 4;           // which 16-lane half
    const int col    = lane & 15;           // particle column / A-matrix row
    const int plocal = wave * 16 + col;     // particle index within block
    const int pglob  = blockIdx.x * PPB + plocal;

    // ---- load initial state into LDS, emit trajectory slice s=0 ----
    for (int idx = tid; idx < PPB * DIM; idx += blockDim.x) {
        int pp = idx >> 5, d = idx & 31;
        int gp = blockIdx.x * PPB + pp;
        float v = x0[(size_t)gp * DIM + d];
        sX[pp * PAD + d] = v;
        traj[((size_t)gp * (NSTEPS + 1)) * DIM + d] = v;
    }
    if (tid < KMIX) {
        float s = 0.f;
        #pragma unroll
        for (int k = 0; k < KMIX; ++k) s += probs[k];
        sLogW[tid] = __logf(probs[tid] / s);
    }
    __syncthreads();

    for (int step = 0; step < NSTEPS; ++step) {
        const float addv = EPS_F * (1.0f - (float)step * DT_F);

        // ---- rebuild per-step mixture tables (fully parallel, 512 entries) ----
        for (int idx = tid; idx < KMIX * DIM; idx += blockDim.x) {
            int k = idx >> 5, d = idx & 31;
            float mu  = mus[idx];
            float var = sigmas[idx] + addv;
            float iv  = 1.0f / var;
            sInvVar[k * PAD + d] = iv;
            sMuInv [k * PAD + d] = mu * iv;
            sTerm  [k * PAD + d] = mu * mu * iv + __logf(var);
        }
        __syncthreads();
        if (tid < KMIX) {
            float c = 0.f;
            #pragma unroll
            for (int d = 0; d < DIM; ++d) c += sTerm[tid * PAD + d];
            sBias[tid] = sLogW[tid] - 0.5f * c;   // softmax-invariant const dropped
        }
        __syncthreads();

        // ---- phase 1: logits^T[k, p] via 16 x v_wmma_f32_16x16x4_f32 ----
        // A[m=k][i]: i even -> -0.5/var at d=i/2 ; i odd -> mu/var at d=i/2
        v2f Areg[16];
        #pragma unroll
        for (int j = 0; j < 16; ++j) {
            int d = 2 * j + half;
            float iv = sInvVar[col * PAD + d];
            Areg[j].x = -0.5f * iv;
            Areg[j].y = sMuInv[col * PAD + d];
        }
        v8f accL;
        #pragma unroll
        for (int r = 0; r < 8; ++r) accL[r] = sBias[r + 8 * half];
        #pragma unroll
        for (int j = 0; j < 16; ++j) {
            float xv = sX[plocal * PAD + 2 * j + half];
            v2f B; B.x = xv * xv; B.y = xv;
            accL = __builtin_amdgcn_wmma_f32_16x16x4_f32(
                false, Areg[j], false, B, (short)0, accL, false, false);
        }

        // ---- phase 2: softmax over 16 components (8 local + cross-half shfl) ----
        float m8 = accL[0];
        #pragma unroll
        for (int r = 1; r < 8; ++r) m8 = fmaxf(m8, accL[r]);
        float m16 = fmaxf(m8, __shfl_xor(m8, 16, 32));
        float w[8], s8 = 0.f;
        #pragma unroll
        for (int r = 0; r < 8; ++r) { w[r] = __expf(accL[r] - m16); s8 += w[r]; }
        float s16 = s8 + __shfl_xor(s8, 16, 32);
        float inv = 1.0f / s16;
        #pragma unroll
        for (int r = 0; r < 8; ++r) w[r] *= inv;
        // other half's 8 weights (same particle column)
        float o[8];
        #pragma unroll
        for (int r = 0; r < 8; ++r) o[r] = __shfl_xor(w[r], 16, 32);

        // drift-GEMM B operands, all statically indexed (8 cndmask on one mask):
        // chunk j needs rows k = 4j+2*half, 4j+1+2*half of W for this particle.
        const bool h = (half != 0);
        v2f Bw[4];
        Bw[0].x = h ? o[2] : w[0];  Bw[0].y = h ? o[3] : w[1];
        Bw[1].x = h ? o[6] : w[4];  Bw[1].y = h ? o[7] : w[5];
        Bw[2].x = h ? w[2] : o[0];  Bw[2].y = h ? w[3] : o[1];
        Bw[3].x = h ? w[6] : o[4];  Bw[3].y = h ? w[7] : o[5];

        // ---- phase 3: drift S1 = invVar^T * W, S2 = (mu/var)^T * W ----
        #pragma unroll
        for (int tp = 0; tp < 2; ++tp) {
            const int dA = 16 * tp + col;          // A-row dim for this lane
            v8f S1; v8f S2;
            #pragma unroll
            for (int r = 0; r < 8; ++r) { S1[r] = 0.f; S2[r] = 0.f; }
            #pragma unroll
            for (int j = 0; j < 4; ++j) {
                int k0 = 4 * j + 2 * half;
                v2f A1; A1.x = sInvVar[k0 * PAD + dA];    A1.y = sInvVar[(k0 + 1) * PAD + dA];
                v2f A2; A2.x = sMuInv [k0 * PAD + dA];    A2.y = sMuInv [(k0 + 1) * PAD + dA];
                S1 = __builtin_amdgcn_wmma_f32_16x16x4_f32(
                    false, A1, false, Bw[j], (short)0, S1, false, false);
                S2 = __builtin_amdgcn_wmma_f32_16x16x4_f32(
                    false, A2, false, Bw[j], (short)0, S2, false, false);
            }
            // lane holds S1/S2 for particle `col`, dims d = 16*tp + 8*half + r
            const int dbase = 16 * tp + 8 * half;
            float out[8];
            #pragma unroll
            for (int rp = 0; rp < 4; ++rp) {
                int d0 = dbase + 2 * rp;
                // unique 26-bit counter: (step+1, particle, dim-pair)
                unsigned int ctr = ((unsigned int)(step + 1) << 19) |
                                   ((unsigned int)pglob << 4) |
                                   (unsigned int)(d0 >> 1);
                unsigned int h1 = lowbias32(ctr);
                unsigned int h2 = lowbias32(ctr ^ 0x9E3779B9u);
                float u1 = (float)(h1 | 1u) * 2.3283064365386963e-10f;   // (0,1]
                float u2 = (float)h2        * 2.3283064365386963e-10f;
                float rr = __fsqrt_rn(-2.0f * __logf(u1));
                float z0 = rr * __cosf(TWO_PI_F * u2);
                float z1 = rr * __sinf(TWO_PI_F * u2);
                float xa = sX[plocal * PAD + d0];
                float xb = sX[plocal * PAD + d0 + 1];
                float dr0 = -EPS_F * (xa * S1[2 * rp]     - S2[2 * rp]);
                float dr1 = -EPS_F * (xb * S1[2 * rp + 1] - S2[2 * rp + 1]);
                out[2 * rp]     = xa + DT_F * dr0 + NOISE_SC * z0;
                out[2 * rp + 1] = xb + DT_F * dr1 + NOISE_SC * z1;
            }
            // update LDS state (each lane owns disjoint dims of its particle)
            #pragma unroll
            for (int r = 0; r < 8; ++r) sX[plocal * PAD + dbase + r] = out[r];
            // trajectory store: 8 consecutive dims -> two float4 (32B aligned)
            size_t off = ((size_t)pglob * (NSTEPS + 1) + (size_t)(step + 1)) * DIM + dbase;
            float4 v0 = make_float4(out[0], out[1], out[2], out[3]);
            float4 v1 = make_float4(out[4], out[5], out[6], out[7]);
            *(float4*)(traj + off)     = v0;
            *(float4*)(traj + off + 4) = v1;
        }
        __syncthreads();   // before next step overwrites the mixture tables
    }
}

__global__ __launch_bounds__(256)
void tout_fill_kernel(float* __restrict__ tout)
{
    size_t i = (size_t)blockIdx.x * blockDim.x + threadIdx.x;
    const size_t total = (size_t)(NSTEPS + 1) * N_PART;
    if (i < total) {
        int s = (int)(i / N_PART);
        tout[i] = (s == 0) ? 0.0f : (float)(s - 1) * DT_F;
    }
}

extern "C" void kernel_launch(void* const* d_in, const int* in_sizes, int n_in,
                              void* d_out, int out_size, void* d_ws, size_t ws_size,
                              hipStream_t stream) {
    const float* x      = (const float*)d_in[0];
    const float* probs  = (const float*)d_in[1];
    const float* mus    = (const float*)d_in[2];
    const float* sigmas = (const float*)d_in[3];

    float* traj = (float*)d_out;
    float* tout = traj + (size_t)N_PART * (NSTEPS + 1) * DIM;

    dim3 grid(N_PART / PPB);        // 256 blocks, 8 wave32 each
    dim3 block(256);
    sde_traj_kernel<<<grid, block, 0, stream>>>(x, probs, mus, sigmas, traj);

    size_t ntout = (size_t)(NSTEPS + 1) * N_PART;
    dim3 g2((unsigned)((ntout + 255) / 256));
    tout_fill_kernel<<<g2, block, 0, stream>>>(tout);
}